// CustomRNN_74947179315400
// MI455X (gfx1250) — compile-verified
//
#include <hip/hip_runtime.h>
#include <math.h>

typedef __attribute__((ext_vector_type(2))) float v2f;
typedef __attribute__((ext_vector_type(4))) float v4f;
typedef __attribute__((ext_vector_type(8))) float v8f;

static constexpr int kD     = 1024;  // hidden size
static constexpr int kNB    = 64;    // sequential blocks
static constexpr int kTiles = 64;    // 16-row tiles per step
static constexpr int kKS    = 16;    // K-splits per step in phase B
static constexpr int kChunk = kD / kKS;  // 64

// workspace layout (bytes)
static constexpr size_t kCntOff   = 0;
static constexpr size_t kHflagOff = (size_t)kNB * kTiles * 4;            // 16 KB
static constexpr size_t kUOff     = 2 * (size_t)kNB * kTiles * 4;        // 32 KB
static constexpr size_t kPartOff  = kUOff + (size_t)kNB * kD * 4;        // +256 KB
static constexpr size_t kCtrlBytes = 2 * (size_t)kNB * kTiles * 4;       // zeroed each call

static __device__ inline v2f vlo(v4f v) { return __builtin_shufflevector(v, v, 0, 1); }
static __device__ inline v2f vhi(v4f v) { return __builtin_shufflevector(v, v, 2, 3); }

static __device__ inline v8f wmma4(v2f a, v2f b, v8f c) {
  return __builtin_amdgcn_wmma_f32_16x16x4_f32(false, a, false, b,
                                               (short)0, c, false, false);
}

// Process 16 memory columns [kb, kb+16) of a 16-row matvec tile with 4 WMMAs.
// Column -> K-slot assignment (global per WMMA, halves load contiguously):
//   wmma0: K0,K1 = kb+0,1   K2,K3 = kb+8,9
//   wmma1: K0,K1 = kb+2,3   K2,K3 = kb+10,11
//   wmma2: K0,K1 = kb+4,5   K2,K3 = kb+12,13
//   wmma3: K0,K1 = kb+6,7   K2,K3 = kb+14,15
// Lane-half 0 (lanes 0-15) loads cols kb+0..7 (one float4 x2), half 1 loads
// kb+8..15 — so A and B are fed by b128 loads instead of b64.
static __device__ inline void wmma_super16(const float* __restrict__ Wrow,
                                           const float* __restrict__ x,
                                           int kb, int koff4, v8f& c0, v8f& c1) {
  v4f a0 = *reinterpret_cast<const v4f*>(Wrow + kb + koff4);
  v4f a1 = *reinterpret_cast<const v4f*>(Wrow + kb + koff4 + 4);
  v4f b0 = *reinterpret_cast<const v4f*>(x + kb + koff4);
  v4f b1 = *reinterpret_cast<const v4f*>(x + kb + koff4 + 4);
  c0 = wmma4(vlo(a0), vlo(b0), c0);
  c1 = wmma4(vhi(a0), vhi(b0), c1);
  c0 = wmma4(vlo(a1), vlo(b1), c0);
  c1 = wmma4(vhi(a1), vhi(b1), c1);
}

// -------- Phase A: U[i] = W_ih[i] @ x[i] + b[i]  (fully parallel) --------
__global__ __launch_bounds__(256) void rnn_phaseA(
    const float* __restrict__ inp, const float* __restrict__ W_ih,
    const float* __restrict__ bias, float* __restrict__ U) {
  const int i    = blockIdx.x >> 3;        // 0..63
  const int tg   = blockIdx.x & 7;
  const int wave = threadIdx.x >> 5;
  const int lane = threadIdx.x & 31;
  const int tile = tg * 8 + wave;          // 0..63
  const int rb   = tile * 16;
  const int koff4 = (lane >> 4) * 8;       // lane-half column offset
  const int row  = rb + (lane & 15);

  const float* Wrow = W_ih + ((size_t)i * kD + row) * kD;
  const float* x    = inp + (size_t)i * kD;

  v8f c0 = {}; v8f c1 = {};
#pragma unroll 4
  for (int kb = 0; kb < kD; kb += 16) {
    __builtin_prefetch(Wrow + kb + 256, 0, 3);   // near-scope streaming prefetch
    wmma_super16(Wrow, x, kb, koff4, c0, c1);
  }
  v8f c = c0 + c1;

  // Column 0 of C lives in lanes 0 (M=0..7) and 16 (M=8..15).
  if (lane == 0) {
#pragma unroll
    for (int r = 0; r < 8; ++r)
      U[(size_t)i * kD + rb + r] = c[r] + bias[(size_t)i * kD + rb + r];
  } else if (lane == 16) {
#pragma unroll
    for (int r = 0; r < 8; ++r)
      U[(size_t)i * kD + rb + 8 + r] = c[r] + bias[(size_t)i * kD + rb + 8 + r];
  }
}

// -------- Phase B: persistent recurrence with tile-granular flag sync --------
// 1024 waves: (tile t, ksplit s). Wave (t,s) at step i waits only on the 4
// h-tiles of step i-1 covering its K chunk, computes a partial matvec with
// WMMA, publishes it, and the (t,0) wave reduces + tanh + releases h tile t.
__global__ __launch_bounds__(256) void rnn_phaseB(
    const float* __restrict__ W_hh, const float* __restrict__ U,
    float* __restrict__ partials, int* cnt, int* hflag,
    float* __restrict__ out) {
  const int wave = threadIdx.x >> 5;
  const int lane = threadIdx.x & 31;
  const int gw   = blockIdx.x * 8 + wave;  // 0..1023
  const int t    = gw >> 4;                // tile 0..63
  const int s    = gw & 15;                // ksplit
  const int rb   = t * 16;
  const int koff4 = (lane >> 4) * 8;
  const int row  = rb + (lane & 15);
  const int kbase = s * kChunk;

  for (int i = 0; i < kNB; ++i) {
    if (i == 0) {
      // h_{-1} == 0, so h_0 = tanh(U[0]); only the reducer wave acts.
      if (s == 0) {
        if (lane < 16) out[rb + lane] = tanhf(U[rb + lane]);
        if (lane == 0)
          __hip_atomic_store(&hflag[t], 1, __ATOMIC_RELEASE,
                             __HIP_MEMORY_SCOPE_AGENT);
      }
      continue;
    }

    // Acquire the 4 h-tiles of step i-1 that cover K chunk [kbase, kbase+64).
#pragma unroll
    for (int q = 0; q < 4; ++q) {
      const int tt = s * 4 + q;
      while (__hip_atomic_load(&hflag[(i - 1) * kTiles + tt], __ATOMIC_ACQUIRE,
                               __HIP_MEMORY_SCOPE_AGENT) == 0)
        __builtin_amdgcn_s_sleep(1);
    }

    const float* x    = out + (size_t)(i - 1) * kD;  // h_{i-1}
    const float* Wrow = W_hh + ((size_t)i * kD + row) * kD;
    v8f c0 = {}; v8f c1 = {};
#pragma unroll
    for (int kb = kbase; kb < kbase + kChunk; kb += 16) {
      __builtin_prefetch(Wrow + kb + 256, 0, 3);
      wmma_super16(Wrow, x, kb, koff4, c0, c1);
    }
    v8f c = c0 + c1;

    float* p = partials + (((size_t)i * kTiles + t) * kKS + s) * 16;
    if (lane == 0) {
#pragma unroll
      for (int r = 0; r < 8; ++r) p[r] = c[r];
    } else if (lane == 16) {
#pragma unroll
      for (int r = 0; r < 8; ++r) p[8 + r] = c[r];
    }
    if (lane == 0)
      __hip_atomic_fetch_add(&cnt[i * kTiles + t], 1, __ATOMIC_RELEASE,
                             __HIP_MEMORY_SCOPE_AGENT);

    if (s == 0) {
      // Reducer: wait for all 16 partials, reduce in a fixed order
      // (deterministic), add U, tanh, publish h tile.
      while (__hip_atomic_load(&cnt[i * kTiles + t], __ATOMIC_ACQUIRE,
                               __HIP_MEMORY_SCOPE_AGENT) < kKS)
        __builtin_amdgcn_s_sleep(1);
      if (lane < 16) {
        float acc = U[(size_t)i * kD + rb + lane];
        const float* pp = partials + (((size_t)i * kTiles + t) * kKS) * 16 + lane;
#pragma unroll
        for (int ss = 0; ss < kKS; ++ss) acc += pp[ss * 16];
        out[(size_t)i * kD + rb + lane] = tanhf(acc);
      }
      if (lane == 0)
        __hip_atomic_store(&hflag[i * kTiles + t], 1, __ATOMIC_RELEASE,
                           __HIP_MEMORY_SCOPE_AGENT);
    }
  }
}

extern "C" void kernel_launch(void* const* d_in, const int* in_sizes, int n_in,
                              void* d_out, int out_size, void* d_ws, size_t ws_size,
                              hipStream_t stream) {
  const float* inp  = (const float*)d_in[0];
  const float* W_ih = (const float*)d_in[1];
  const float* W_hh = (const float*)d_in[2];
  const float* bias = (const float*)d_in[3];
  float* out = (float*)d_out;

  char* ws = (char*)d_ws;
  int*   cnt      = (int*)(ws + kCntOff);
  int*   hflag    = (int*)(ws + kHflagOff);
  float* U        = (float*)(ws + kUOff);
  float* partials = (float*)(ws + kPartOff);

  // Flags/counters must start at zero every call (ws is not re-poisoned).
  hipMemsetAsync(d_ws, 0, kCtrlBytes, stream);

  rnn_phaseA<<<kNB * 8, 256, 0, stream>>>(inp, W_ih, bias, U);
  rnn_phaseB<<<128, 256, 0, stream>>>(W_hh, U, partials, cnt, hflag, out);
}